// DynamicWeightProjection_66365834657872
// MI455X (gfx1250) — compile-verified
//
#include <hip/hip_runtime.h>
#include <hip/hip_bf16.h>
#include <math.h>

// ---------------------------------------------------------------------------
// Problem constants (B=2, T=2048, H=32, DK=128 -> D=4096; G=1, C=4, K=128,
// I2=64, M=32).  BT = B*T = 4096 tokens.
// ---------------------------------------------------------------------------
constexpr int BT   = 4096;   // B*T
constexpr int DDIM = 4096;   // H*DK
constexpr int GCK  = 512;    // G*C*K
constexpr int KC   = 128;    // K per c-slice
constexpr int N2   = 2048;   // I2*M per c-slice

// Output offsets (floats) in return order.
constexpr size_t SZ_QW = (size_t)BT * 1024;  // (B,T,1,32,32)
constexpr size_t SZ_DD = (size_t)BT * 32;    // (B,T,1,32)
constexpr size_t OFF_PRE_QW1  = 0;
constexpr size_t OFF_PRE_QW2  = OFF_PRE_QW1  + SZ_QW;
constexpr size_t OFF_PRE_KW1  = OFF_PRE_QW2  + SZ_QW;
constexpr size_t OFF_PRE_KW2  = OFF_PRE_KW1  + SZ_QW;
constexpr size_t OFF_PRE_QDD  = OFF_PRE_KW2  + SZ_QW;
constexpr size_t OFF_PRE_KDD  = OFF_PRE_QDD  + SZ_DD;
constexpr size_t OFF_POST_QW1 = OFF_PRE_KDD  + SZ_DD;
constexpr size_t OFF_POST_QW2 = OFF_POST_QW1 + SZ_QW;
constexpr size_t OFF_POST_KW1 = OFF_POST_QW2 + SZ_QW;
constexpr size_t OFF_POST_KW2 = OFF_POST_KW1 + SZ_QW;
constexpr size_t OFF_POST_QDD = OFF_POST_KW2 + SZ_QW;
constexpr size_t OFF_POST_KDD = OFF_POST_QDD + SZ_DD;
constexpr size_t OFF_KW       = OFF_POST_KDD + SZ_DD;  // (B,T,2,32,32)

typedef __attribute__((ext_vector_type(2))) float v2f;
typedef __attribute__((ext_vector_type(8))) float v8f;

__device__ __forceinline__ v8f wmma_f32x4(v2f a, v2f b, v8f c) {
  // V_WMMA_F32_16X16X4_F32 : D = A(16x4) * B(4x16) + C(16x16)
  return __builtin_amdgcn_wmma_f32_16x16x4_f32(false, a, false, b, (short)0, c,
                                               false, false);
}

// ---------------------------------------------------------------------------
// Tensor Data Mover: async DMA of a 2-D tile (32-bit elements) into LDS.
// D# packing per CDNA5 ISA 8.3/8.4: group0={count,lds_addr,global_addr,type=2}
// group1={data_size, pad ctl, tensor dims, tile dims, dim0 stride}; groups
// 2/3 (and the trailing group slot) zero for a 2-D tensor.  One DMA per
// *wave* (EXEC ignored).  This toolchain exposes the 6-arg builtin:
//   (uint32x4 g0, int32x8 g1, int32x4, int32x4, int32x8, i32 cpol)
// ---------------------------------------------------------------------------
#if defined(__HIP_DEVICE_COMPILE__) && __has_builtin(__builtin_amdgcn_tensor_load_to_lds)
#define USE_TDM 1
#else
#define USE_TDM 0
#endif

#if USE_TDM
typedef unsigned int v4u __attribute__((ext_vector_type(4)));
typedef int v8i __attribute__((ext_vector_type(8)));
typedef int v4i __attribute__((ext_vector_type(4)));

__device__ __forceinline__ void tdm_load_2d(unsigned int lds_byte_off,
                                            const float* gptr,
                                            unsigned int tensor_d0,  // elems
                                            unsigned int tensor_d1,  // rows
                                            unsigned int stride_d0,  // elems
                                            unsigned int tile_d0,
                                            unsigned int tile_d1,
                                            bool pad_row) {
  unsigned long long ga = (unsigned long long)(uintptr_t)gptr;
  v4u g0;
  g0[0] = 1u;                                   // count=1, user mode
  g0[1] = lds_byte_off;                         // lds_addr (bytes)
  g0[2] = (unsigned int)(ga & 0xffffffffu);     // global_addr[31:0]
  g0[3] = (unsigned int)((ga >> 32) & 0x1ffffffu) | (2u << 30);  // [56:32]+type2
  v8i g1;
  // data_size=2 (4B).  Optional pad: 1 dword after every 32 dwords
  // (pad_interval code 4 = 32 dwords, pad_amount code 0 = 1 dword).
  unsigned int w0 = (2u << 16);
  if (pad_row) w0 |= (1u << 20) | (4u << 22);   // pad_enable, interval=32dw
  g1[0] = (int)w0;
  g1[1] = (int)((tensor_d0 & 0xffffu) << 16);                       // dim0 lo
  g1[2] = (int)(((tensor_d0 >> 16) & 0xffffu) |
                ((tensor_d1 & 0xffffu) << 16));                     // dim0hi|dim1lo
  g1[3] = (int)(((tensor_d1 >> 16) & 0xffffu) |
                ((tile_d0 & 0xffffu) << 16));                       // dim1hi|tile0
  g1[4] = (int)(tile_d1 & 0xffffu);                                 // tile1, tile2=0
  g1[5] = (int)stride_d0;                                           // stride[31:0]
  g1[6] = 0;                                                        // stride hi, d1stride
  g1[7] = 0;
  v4i z4 = {0, 0, 0, 0};
  v8i z8 = {0, 0, 0, 0, 0, 0, 0, 0};
  __builtin_amdgcn_tensor_load_to_lds(g0, g1, z4, z4, z8, 0);
}
#endif

// ---------------------------------------------------------------------------
// Tiled f32 WMMA GEMM.  Block: 256 threads (8 waves), tile 128x128, K-step 32.
// Wave (wm 0..1, wn 0..3) owns a 64x32 sub-tile = 4x2 WMMA accumulators.
// A tile (128x32) staged in LDS with 33-float row stride (TDM row pad / manual
// pad) to kill bank conflicts on strided A-fragment reads; B tile (32x128)
// packed.  TDM path double-buffers: issue stage k+1, s_wait_tensorcnt<=2 for
// stage k, barrier, compute.
// EPI: 0 none, 1 gelu, 2 tanh, 3 scatter into w1/w2 output slots.
// ---------------------------------------------------------------------------
constexpr int AS_STRIDE = 33;                // padded A row stride (floats)
constexpr int AS_TILE   = 128 * AS_STRIDE;   // 4224
constexpr int BS_TILE   = 32 * 128;          // 4096

template <int EPI>
__global__ __launch_bounds__(256) void wmma_gemm_tiled(
    const float* __restrict__ A, const float* __restrict__ B,
    float* __restrict__ Cout, float* __restrict__ outW1,
    float* __restrict__ outW2, int N, int K, int lda, int ldb, int ldc) {
  __shared__ float lds[2 * AS_TILE + 2 * BS_TILE];  // ~66 KB
  const int AS[2] = {0, AS_TILE};
  const int BS[2] = {2 * AS_TILE, 2 * AS_TILE + BS_TILE};

  const int tid  = threadIdx.x;
  const int lane = tid & 31;
  const int wave = tid >> 5;
  const int wm   = wave >> 2;  // 0..1
  const int wn   = wave & 3;   // 0..3
  const int half = lane >> 4;
  const int l15  = lane & 15;

  const int bm0 = blockIdx.y * 128;
  const int bn0 = blockIdx.x * 128;
  const int tmw = wm * 64;
  const int tnw = wn * 32;

  v8f acc[4][2];
#pragma unroll
  for (int mi = 0; mi < 4; ++mi)
#pragma unroll
    for (int ni = 0; ni < 2; ++ni) acc[mi][ni] = (v8f)(0.0f);

  const int nkb = K >> 5;  // K / 32

#if USE_TDM
  if (wave == 0) {
    tdm_load_2d((unsigned)(AS[0] * 4), A + (size_t)bm0 * lda,
                (unsigned)K, 128u, (unsigned)lda, 32u, 128u, true);
    tdm_load_2d((unsigned)(BS[0] * 4), B + bn0,
                (unsigned)(N - bn0), 32u, (unsigned)ldb, 128u, 32u, false);
  }
#endif

  for (int kb = 0; kb < nkb; ++kb) {
    const int cur = kb & 1;
#if USE_TDM
    if (wave == 0) {
      if (kb + 1 < nkb) {
        const int nk0 = (kb + 1) * 32;
        tdm_load_2d((unsigned)(AS[cur ^ 1] * 4),
                    A + (size_t)bm0 * lda + nk0,
                    (unsigned)(K - nk0), 128u, (unsigned)lda, 32u, 128u, true);
        tdm_load_2d((unsigned)(BS[cur ^ 1] * 4),
                    B + (size_t)nk0 * ldb + bn0,
                    (unsigned)(N - bn0), 32u, (unsigned)ldb, 128u, 32u, false);
        __builtin_amdgcn_s_wait_tensorcnt(2);  // stage kb complete (in-order)
      } else {
        __builtin_amdgcn_s_wait_tensorcnt(0);
      }
    }
    __syncthreads();  // LDS stage kb visible to all waves
#else
    // Fallback: cooperative global->LDS staging (single buffer).
    __syncthreads();
    {
      const float* gA = A + (size_t)bm0 * lda + kb * 32;
      const float* gB = B + (size_t)(kb * 32) * ldb + bn0;
      for (int f = tid; f < 1024; f += 256) {  // A: 128x32 = 1024 float4/4
        const int row = f >> 3, c4 = (f & 7) * 4;
        const float4 t = *(const float4*)(gA + (size_t)row * lda + c4);
        float* d = lds + AS[cur] + row * AS_STRIDE + c4;
        d[0] = t.x; d[1] = t.y; d[2] = t.z; d[3] = t.w;
      }
      for (int f = tid; f < 1024; f += 256) {  // B: 32x128
        const int row = f >> 5, c4 = (f & 31) * 4;
        const float4 t = *(const float4*)(gB + (size_t)row * ldb + c4);
        float* d = lds + BS[cur] + row * 128 + c4;
        d[0] = t.x; d[1] = t.y; d[2] = t.z; d[3] = t.w;
      }
    }
    __syncthreads();
#endif

    const float* As = lds + AS[cur];
    const float* Bs = lds + BS[cur];
#pragma unroll
    for (int kk = 0; kk < 8; ++kk) {
      const int kb4 = kk * 4 + 2 * half;
      v2f a[4], b[2];
#pragma unroll
      for (int mi = 0; mi < 4; ++mi) {
        const int row = tmw + mi * 16 + l15;
        a[mi].x = As[row * AS_STRIDE + kb4];
        a[mi].y = As[row * AS_STRIDE + kb4 + 1];
      }
#pragma unroll
      for (int ni = 0; ni < 2; ++ni) {
        const int col = tnw + ni * 16 + l15;
        b[ni].x = Bs[kb4 * 128 + col];
        b[ni].y = Bs[(kb4 + 1) * 128 + col];
      }
#pragma unroll
      for (int mi = 0; mi < 4; ++mi)
#pragma unroll
        for (int ni = 0; ni < 2; ++ni)
          acc[mi][ni] = wmma_f32x4(a[mi], b[ni], acc[mi][ni]);
    }
    __syncthreads();  // done reading stage kb before its buffer is re-filled
  }

  // Epilogue
#pragma unroll
  for (int mi = 0; mi < 4; ++mi)
#pragma unroll
    for (int ni = 0; ni < 2; ++ni)
#pragma unroll
      for (int j = 0; j < 8; ++j) {
        const int m = bm0 + tmw + mi * 16 + j + 8 * half;
        const int n = bn0 + tnw + ni * 16 + l15;
        float v = acc[mi][ni][j];
        if (EPI == 1) v = 0.5f * v * (1.0f + erff(v * 0.70710678118654752440f));
        if (EPI == 2) v = tanhf(v);
        if (EPI == 3) {
          const int i  = n >> 5;   // I2 index 0..63
          const int mm = n & 31;   // M index
          if (i < 32)
            outW1[(size_t)m * 1024 + i * 32 + mm] = v;
          else
            outW2[(size_t)m * 1024 + (i - 32) * 32 + mm] = v;
        } else {
          Cout[(size_t)m * ldc + n] = v;
        }
      }
}

// ---------------------------------------------------------------------------
// Per-token finalize (one block / token, 256 threads = 8 waves):
//  * tid<128  : rms-normalize the 4x32 w1 rows in place; stash kw1 (c=1,3)
//               normalized rows in LDS for the KW GEMM.
//  * tid>=128 : scatter tanh'd ddv into pre/post qdd/kdd.
//  * waves 0,1: KW[p] = kw1n^T @ kw2 + diag(kdd)  via 2x2 tiles of
//               V_WMMA_F32_16X16X4_F32 (32 WMMA / token / matrix).
// ---------------------------------------------------------------------------
__global__ __launch_bounds__(256) void finalize_kernel(
    const float* __restrict__ ddv,  // BT x 128 (tanh already applied)
    float* __restrict__ out) {
  __shared__ float lds_kw1[2][32 * 32];  // [p][i*32+m] normalized kw1

  const int t    = blockIdx.x;
  const int tid  = threadIdx.x;
  const int lane = tid & 31;
  const int wave = tid >> 5;

  if (tid < 128) {
    const size_t base1[4] = {OFF_PRE_QW1, OFF_PRE_KW1, OFF_POST_QW1,
                             OFF_POST_KW1};
    const int c = tid >> 5;  // 0..3
    const int i = tid & 31;  // 0..31
    float* row = out + base1[c] + (size_t)t * 1024 + (size_t)i * 32;
    float v[32];
    float ss = 0.0f;
#pragma unroll
    for (int m = 0; m < 32; ++m) {
      v[m] = row[m];
      ss += v[m] * v[m];
    }
    const float s = rsqrtf(ss * (1.0f / 32.0f) + 1e-6f);
    const int stash = (c == 1) ? 0 : ((c == 3) ? 1 : -1);
#pragma unroll
    for (int m = 0; m < 32; ++m) {
      const float nv = v[m] * s;
      row[m] = nv;
      if (stash >= 0) lds_kw1[stash][i * 32 + m] = nv;
    }
  } else {
    const size_t ddbase[4] = {OFF_PRE_QDD, OFF_PRE_KDD, OFF_POST_QDD,
                              OFF_POST_KDD};
    const int j = tid - 128;  // 0..127
    const float v = ddv[(size_t)t * 128 + j];
    out[ddbase[j >> 5] + (size_t)t * 32 + (j & 31)] = v;
  }

  __syncthreads();

  if (wave < 2) {
    const int p    = wave;  // 0 = pre (c=1), 1 = post (c=3)
    const int half = lane >> 4;
    const int l15  = lane & 15;
    const float* kw1n = lds_kw1[p];  // [k][m] normalized kw1, k = I index
    const float* kw2 =
        out + (p ? OFF_POST_KW2 : OFF_PRE_KW2) + (size_t)t * 1024;  // [k][n]

    v8f acc[2][2];
#pragma unroll
    for (int mt = 0; mt < 2; ++mt)
#pragma unroll
      for (int nt = 0; nt < 2; ++nt) acc[mt][nt] = (v8f)(0.0f);

#pragma unroll
    for (int kk = 0; kk < 8; ++kk) {
      const int kb = kk * 4 + 2 * half;
      v2f a0, a1, b0, b1;
      a0.x = kw1n[kb * 32 + l15];          // A = kw1n^T : A[m][k]=kw1n[k][m]
      a0.y = kw1n[(kb + 1) * 32 + l15];
      a1.x = kw1n[kb * 32 + 16 + l15];
      a1.y = kw1n[(kb + 1) * 32 + 16 + l15];
      b0.x = kw2[kb * 32 + l15];
      b0.y = kw2[(kb + 1) * 32 + l15];
      b1.x = kw2[kb * 32 + 16 + l15];
      b1.y = kw2[(kb + 1) * 32 + 16 + l15];
      acc[0][0] = wmma_f32x4(a0, b0, acc[0][0]);
      acc[0][1] = wmma_f32x4(a0, b1, acc[0][1]);
      acc[1][0] = wmma_f32x4(a1, b0, acc[1][0]);
      acc[1][1] = wmma_f32x4(a1, b1, acc[1][1]);
    }

    const float* kdd = ddv + (size_t)t * 128 + (p ? 96 : 32);
    float* kwout = out + OFF_KW + ((size_t)t * 2 + p) * 1024;
#pragma unroll
    for (int mt = 0; mt < 2; ++mt)
#pragma unroll
      for (int nt = 0; nt < 2; ++nt)
#pragma unroll
        for (int j = 0; j < 8; ++j) {
          const int m = mt * 16 + j + 8 * half;
          const int n = nt * 16 + l15;
          float v = acc[mt][nt][j];
          if (m == n) v += kdd[m];  // + kdd * I
          kwout[m * 32 + n] = v;
        }
  }
}

// ---------------------------------------------------------------------------
extern "C" void kernel_launch(void* const* d_in, const int* in_sizes, int n_in,
                              void* d_out, int out_size, void* d_ws,
                              size_t ws_size, hipStream_t stream) {
  const float* qv  = (const float*)d_in[0];  // (B,T,H,DK) == (BT, D)
  const float* dw1 = (const float*)d_in[1];  // (D, G*C*K) = 4096 x 512
  const float* qkw = (const float*)d_in[2];  // (C, 128, 2048)
  const float* dd  = (const float*)d_in[3];  // (D, 128)
  float* out = (float*)d_out;

  float* hidden = (float*)d_ws;               // 4096 x 512  (8 MB)
  float* ddv    = hidden + (size_t)BT * GCK;  // 4096 x 128  (2 MB)

  // 1) hidden = gelu(qv @ dw1)   [4096x4096]x[4096x512], TDM-fed tiles
  wmma_gemm_tiled<1><<<dim3(GCK / 128, BT / 128), 256, 0, stream>>>(
      qv, dw1, hidden, nullptr, nullptr, GCK, DDIM, DDIM, GCK, GCK);

  // 2) ddv = tanh(qv @ dd)       [4096x4096]x[4096x128]
  wmma_gemm_tiled<2><<<dim3(1, BT / 128), 256, 0, stream>>>(
      qv, dd, ddv, nullptr, nullptr, 128, DDIM, DDIM, 128, 128);

  // 3) w12_c scattered straight into out: 4 GEMMs [4096x128]x[128x2048]
  {
    const size_t b1[4] = {OFF_PRE_QW1, OFF_PRE_KW1, OFF_POST_QW1, OFF_POST_KW1};
    const size_t b2[4] = {OFF_PRE_QW2, OFF_PRE_KW2, OFF_POST_QW2, OFF_POST_KW2};
    for (int c = 0; c < 4; ++c) {
      wmma_gemm_tiled<3><<<dim3(N2 / 128, BT / 128), 256, 0, stream>>>(
          hidden + c * KC, qkw + (size_t)c * KC * N2, nullptr, out + b1[c],
          out + b2[c], N2, KC, GCK, N2, 0);
    }
  }

  // 4) per-token rms-norm (in place), ddv scatter, KW = kw1n^T kw2 + diag
  finalize_kernel<<<BT, 256, 0, stream>>>(ddv, out);
}